// OTLoss_60919816126917
// MI455X (gfx1250) — compile-verified
//
#include <hip/hip_runtime.h>

// ---------------- problem constants ----------------
#define INPUT_SIZE 448
#define OUTD 56
#define OUTP 64
#define NPTS 512
#define BATCH 32
#define NITER 100
#define M2 (OUTD*OUTD)          // 3136

// ---------------- LDS layout (floats) ----------------
// pitches chosen so tile accesses are bank-conflict free and 8B aligned
#define KX_PITCH 66             // Kx[n][j], n=0..511, j=0..63 (padded, zero)
#define KYT_PITCH 514           // KyT[i][n], i=0..63 (padded, zero)
#define V_PITCH 66              // V[i][j] 64x64 region used, padded zero

#define KX_OFF  0
#define KX_SZ   (NPTS*KX_PITCH)          // 33792
#define KYT_OFF (KX_OFF + KX_SZ)
#define KYT_SZ  (OUTP*KYT_PITCH)         // 32896
#define V_OFF   (KYT_OFF + KYT_SZ)
#define V_SZ    (OUTP*V_PITCH)           // 4224
#define B_OFF   (V_OFF + V_SZ)
#define B_SZ    M2                       // 3136
#define U_OFF   (B_OFF + B_SZ)
#define U_SZ    NPTS                     // 512
#define WP_OFF  (U_OFF + U_SZ)
#define WP_SZ   (8*NPTS)                 // 4096 (also reused for beta)
#define SMEM_FLOATS (WP_OFF + WP_SZ)     // 78656 floats = 314,624 bytes < 320KB

typedef float v2f __attribute__((ext_vector_type(2)));
typedef float v8f __attribute__((ext_vector_type(8)));

__device__ __forceinline__ float block_sum(float v, float* sred, int tid) {
    #pragma unroll
    for (int off = 16; off > 0; off >>= 1) v += __shfl_down(v, off, 32);
    __syncthreads();                       // protect sred reuse across calls
    if ((tid & 31) == 0) sred[tid >> 5] = v;
    __syncthreads();
    float t = (tid < 16) ? sred[tid] : 0.0f;
    #pragma unroll
    for (int off = 8; off > 0; off >>= 1) t += __shfl_down(t, off, 32);
    if (tid == 0) sred[0] = t;
    __syncthreads();
    return sred[0];
}

__global__ __launch_bounds__(512, 1)
void ot_sinkhorn(const float* __restrict__ pred,
                 const float* __restrict__ normed,
                 const float* __restrict__ pts,
                 float* __restrict__ ws)
{
    extern __shared__ float sm[];
    float* Kx  = sm + KX_OFF;   // [n][j]
    float* KyT = sm + KYT_OFF;  // [i][n]
    float* V   = sm + V_OFF;    // [i][j]
    float* Bm  = sm + B_OFF;    // b (target density), i*56+j
    float* U   = sm + U_OFF;    // u[n]
    float* WP  = sm + WP_OFF;   // per-(i-block,half) Kv partials / later beta
    __shared__ float sred[16];

    const int tid  = threadIdx.x;
    const int s    = blockIdx.x;
    const int lane = tid & 31;
    const int wv   = tid >> 5;          // 16 waves
    const int half = lane >> 4;         // K-half of the WMMA tile
    const int l    = lane & 15;

    // ---------------- precompute Kx, KyT, b, u0 ----------------
    const int   n  = tid;               // exactly 512 threads
    const float px = pts[(s*NPTS + n)*2 + 0];
    const float py = pts[(s*NPTS + n)*2 + 1];
    const float x  = px * (2.0f/INPUT_SIZE) - 1.0f;
    const float y  = py * (2.0f/INPUT_SIZE) - 1.0f;
    for (int j = 0; j < OUTP; ++j) {
        float kx = 0.0f, ky = 0.0f;
        if (j < OUTD) {
            float c  = (float)j*(16.0f/INPUT_SIZE) + (8.0f/INPUT_SIZE) - 1.0f;
            float dx = x - c, dy = y - c;
            kx = __expf(-dx*dx*0.1f);   // exp(-dist/REG), REG=10
            ky = __expf(-dy*dy*0.1f);
        }
        Kx[n*KX_PITCH + j]   = kx;
        KyT[j*KYT_PITCH + n] = ky;
    }
    U[n] = 1.0f/(float)NPTS;
    for (int idx = tid; idx < M2;   idx += 512) Bm[idx] = normed[s*M2 + idx];
    for (int idx = tid; idx < V_SZ; idx += 512) V[idx]  = 0.0f;  // zero padding rows/cols
    __syncthreads();

    // ---------------- Sinkhorn iterations, all in LDS ----------------
    for (int it = 0; it < NITER; ++it) {
        // GEMM1: S[i,j] = sum_n (u[n]*Ky[n,i]) * Kx[n,j]; then v = b/(S+1e-16)
        {
            const int iblock = (wv >> 2) * 16;
            const int jblock = (wv & 3) * 16;
            v8f acc = {0.f,0.f,0.f,0.f,0.f,0.f,0.f,0.f};
            const float* kyrow = &KyT[(iblock + l)*KYT_PITCH];
            const float* kxcol = &Kx[jblock + l];
            for (int n0 = 0; n0 < NPTS; n0 += 4) {
                const int na = n0 + 2*half;
                v2f kyv = *(const v2f*)(kyrow + na);            // ds_load_b64
                v2f a   = { kyv.x * U[na], kyv.y * U[na+1] };   // A[M=i,K=n]
                v2f b   = { kxcol[na*KX_PITCH], kxcol[(na+1)*KX_PITCH] }; // B[K=n,N=j]
                acc = __builtin_amdgcn_wmma_f32_16x16x4_f32(
                        false, a, false, b, (short)0, acc, false, false);
            }
            #pragma unroll
            for (int k = 0; k < 8; ++k) {
                const int i = iblock + k + 8*half;
                const int j = jblock + l;
                if (i < OUTD && j < OUTD)
                    V[i*V_PITCH + j] = Bm[i*OUTD + j] / (acc[k] + 1e-16f);
            }
        }
        __syncthreads();

        // GEMM2: W'[i,n] = sum_j V[i,j]*Kx[n,j]; fold with Ky into Kv partials
        for (int t8 = 0; t8 < 8; ++t8) {
            const int t      = wv*8 + t8;           // 128 tiles total
            const int iblock = (t & 3)  * 16;
            const int nblock = (t >> 2) * 16;
            v8f acc = {0.f,0.f,0.f,0.f,0.f,0.f,0.f,0.f};
            const float* vrow  = &V[(iblock + l)*V_PITCH];
            const float* kxrow = &Kx[(nblock + l)*KX_PITCH];
            for (int j0 = 0; j0 < OUTD; j0 += 4) {
                const int ja = j0 + 2*half;
                v2f a = *(const v2f*)(vrow  + ja);  // A[M=i,K=j]
                v2f b = *(const v2f*)(kxrow + ja);  // B[K=j,N=n]
                acc = __builtin_amdgcn_wmma_f32_16x16x4_f32(
                        false, a, false, b, (short)0, acc, false, false);
            }
            const int nn = nblock + l;
            float ssum = 0.0f;
            #pragma unroll
            for (int k = 0; k < 8; ++k) {
                const int i = iblock + k + 8*half;
                ssum += KyT[i*KYT_PITCH + nn] * acc[k];
            }
            WP[((iblock >> 3) + half)*NPTS + nn] = ssum;  // p = 2*(iblock/16)+half
        }
        __syncthreads();

        // u update (deterministic fixed-order reduce of 8 partials)
        {
            float kv = 0.0f;
            #pragma unroll
            for (int p = 0; p < 8; ++p) kv += WP[p*NPTS + n];
            U[n] = (1.0f/(float)NPTS) / (kv + 1e-16f);
        }
        __syncthreads();
    }

    // ---------------- epilogue: beta, ot_obj, loss, wd ----------------
    const float* predp = pred + s*M2;
    float* Beta = WP;                       // reuse scratch
    float ot_p = 0.f, sc_p = 0.f, spb_p = 0.f;
    for (int idx = tid; idx < M2; idx += 512) {
        const int i = idx / OUTD, j = idx - i*OUTD;
        const float vv   = V[i*V_PITCH + j];
        const float beta = 10.0f * __logf(vv + 1e-16f);
        Beta[idx] = beta;
        const float p = predp[idx];
        ot_p  += Bm[idx] * beta;
        sc_p  += p;
        spb_p += p * beta;
    }
    __syncthreads();
    const float ot    = block_sum(ot_p,  sred, tid);
    const float sc    = block_sum(sc_p,  sred, tid);
    const float spb   = block_sum(spb_p, sred, tid);
    const float denom = sc*sc + 1e-8f;
    const float gs = sc / denom, go = spb / denom;

    float loss_p = 0.0f;
    for (int idx = tid; idx < M2; idx += 512)
        loss_p += predp[idx] * (gs * Beta[idx] - go);
    const float loss = block_sum(loss_p, sred, tid);

    // wd = sum_n u[n] * sum_i Ky[n,i] * ( yd[n,i]*sum_j Kx*V + sum_j xd*Kx*V )
    float wd_p = 0.0f;
    {
        const float un = U[n];
        for (int i = 0; i < OUTD; ++i) {
            const float ci  = (float)i*(16.0f/INPUT_SIZE) + (8.0f/INPUT_SIZE) - 1.0f;
            const float dy  = y - ci;
            const float yd  = dy*dy;
            const float kyi = KyT[i*KYT_PITCH + n];
            float s1 = 0.f, s2 = 0.f;
            for (int j = 0; j < OUTD; ++j) {
                const float cj = (float)j*(16.0f/INPUT_SIZE) + (8.0f/INPUT_SIZE) - 1.0f;
                const float dx = x - cj;
                const float kv = Kx[n*KX_PITCH + j] * V[i*V_PITCH + j];
                s1 += kv;
                s2 += dx*dx*kv;
            }
            wd_p += kyi * (yd*s1 + s2);
        }
        wd_p *= un;
    }
    const float wd = block_sum(wd_p, sred, tid);

    if (tid == 0) {
        ws[3*s + 0] = loss;
        ws[3*s + 1] = wd;
        ws[3*s + 2] = ot;
    }
}

// deterministic fixed-order final reduction over samples
__global__ void ot_reduce(const float* __restrict__ ws, float* __restrict__ out) {
    if (threadIdx.x == 0 && blockIdx.x == 0) {
        float l = 0.f, w = 0.f, o = 0.f;
        for (int s2 = 0; s2 < BATCH; ++s2) {
            l += ws[3*s2 + 0];
            w += ws[3*s2 + 1];
            o += ws[3*s2 + 2];
        }
        out[0] = l;   // loss
        out[1] = w;   // wd
        out[2] = o;   // ot_obj
    }
}

extern "C" void kernel_launch(void* const* d_in, const int* in_sizes, int n_in,
                              void* d_out, int out_size, void* d_ws, size_t ws_size,
                              hipStream_t stream) {
    (void)in_sizes; (void)n_in; (void)out_size; (void)ws_size;
    const float* pred   = (const float*)d_in[0];  // pred_density
    const float* normed = (const float*)d_in[1];  // normed_pred_density
    const float* pts    = (const float*)d_in[2];  // target_points
    float* out = (float*)d_out;
    float* ws  = (float*)d_ws;

    const size_t smem = SMEM_FLOATS * sizeof(float);  // ~314.6 KB < 320 KB/WGP
    ot_sinkhorn<<<BATCH, 512, smem, stream>>>(pred, normed, pts, ws);
    ot_reduce<<<1, 32, 0, stream>>>(ws, out);
}